// TrustSVDModel_35296041238739
// MI455X (gfx1250) — compile-verified
//
#include <hip/hip_runtime.h>
#include <hip/hip_bf16.h>

#define LAMDA   0.5f
#define LAMDA_T 0.25f
#define DD 64

typedef __attribute__((ext_vector_type(2))) float v2f;
typedef __attribute__((ext_vector_type(8))) float v8f;

#ifndef __has_builtin
#define __has_builtin(x) 0
#endif
#if __has_builtin(__builtin_amdgcn_wmma_f32_16x16x4_f32)
#define HAVE_WMMA_F32_4 1
#else
#define HAVE_WMMA_F32_4 0
#endif

// ---------------------------------------------------------------- reductions

__device__ __forceinline__ float block_reduce_add(float v, float* sdata) {
  int lane = threadIdx.x & 31;
  int wave = threadIdx.x >> 5;
#pragma unroll
  for (int m = 16; m >= 1; m >>= 1) v += __shfl_xor(v, m, 32);
  if (lane == 0) sdata[wave] = v;
  __syncthreads();
  float r = 0.0f;
  if (wave == 0) {
    r = (lane < (int)(blockDim.x >> 5)) ? sdata[lane] : 0.0f;
#pragma unroll
    for (int m = 4; m >= 1; m >>= 1) r += __shfl_xor(r, m, 32);
  }
  return r;  // valid on wave 0, lane 0
}

// ------------------------------------------- 16 batched dot products / wave
// Computes dot(A[sidx[tileBase+m]], B[didx[tileBase+m]]) for m = 0..15 using
// V_WMMA_F32_16X16X4_F32 (diagonal of the 16x16 cross product). Full f32.

struct Dot16 {
  float val;    // dot product for edge tileBase + m (on active lanes)
  int   m;      // which edge within the tile this lane holds
  int   ra, rb; // row indices this lane gathered (match m on active lanes)
  bool  active; // exactly 16 lanes active, distinct m
};

__device__ __forceinline__ Dot16 tile_dot16(const float* __restrict__ A,
                                            const float* __restrict__ B,
                                            const int* __restrict__ sidx,
                                            const int* __restrict__ didx,
                                            int tileBase, int nEdges) {
  Dot16 r;
  int lane = threadIdx.x & 31;
#if HAVE_WMMA_F32_4
  int m = lane & 15;
  int e = tileBase + m;
  if (e >= nEdges) e = nEdges - 1;            // clamp, keep EXEC full for WMMA
  int ra = sidx[e], rb = didx[e];
  int kOff = (lane >= 16) ? 2 : 0;            // lanes 0-15: K={0,1}; 16-31: K={2,3}
  const float* pa = A + (size_t)ra * DD + kOff;
  const float* pb = B + (size_t)rb * DD + kOff;
  v8f c = {};
#pragma unroll
  for (int kb = 0; kb < DD; kb += 4) {
    v2f a = *(const v2f*)(pa + kb);           // 8B-aligned float2 loads
    v2f b = *(const v2f*)(pb + kb);
    c = __builtin_amdgcn_wmma_f32_16x16x4_f32(false, a, false, b,
                                              (short)0, c, false, false);
  }
  // diagonal extraction from C/D layout (VGPR r: lanes0-15 M=r, lanes16-31 M=r+8)
  bool act; int om, ci;
  if (lane < 8)        { act = true;  om = lane;      ci = lane; }
  else if (lane >= 24) { act = true;  om = lane - 16; ci = lane - 24; }
  else                 { act = false; om = 0;         ci = 0; }
  float v = c[0];
#pragma unroll
  for (int q = 1; q < 8; ++q)
    if (ci == q) v = c[q];
  r.val = v; r.m = om; r.active = act; r.ra = ra; r.rb = rb;
#else
  // fallback: lanes 0..15 each compute their dot product serially
  int m = lane & 15;
  int e = tileBase + m;
  if (e >= nEdges) e = nEdges - 1;
  int ra = sidx[e], rb = didx[e];
  const float* pa = A + (size_t)ra * DD;
  const float* pb = B + (size_t)rb * DD;
  float v = 0.0f;
  for (int k = 0; k < DD; ++k) v += pa[k] * pb[k];
  r.val = v; r.m = m; r.active = (lane < 16); r.ra = ra; r.rb = rb;
#endif
  return r;
}

// ------------------------------------------------------------------ kernels

__global__ void deg2_kernel(const int* __restrict__ ia, float* __restrict__ da,
                            const int* __restrict__ ib, float* __restrict__ db, int n) {
  int e = blockIdx.x * blockDim.x + threadIdx.x;
  if (e < n) {
    atomicAdd(&da[ia[e]], 1.0f);
    atomicAdd(&db[ib[e]], 1.0f);
  }
}

__global__ void factor_kernel(float* __restrict__ f, int n) {
  int i = blockIdx.x * blockDim.x + threadIdx.x;
  if (i < n) {
    float deg = f[i];
    f[i] = (deg > 0.0f) ? (1.0f / sqrtf(deg)) : 0.0f;
  }
}

__global__ void copy4_kernel(const float4* __restrict__ src, float4* __restrict__ dst, int n4) {
  int i = blockIdx.x * blockDim.x + threadIdx.x;
  if (i < n4) dst[i] = src[i];
}

// h_user[seg[e]] += fac[seg[e]] * tab[gat[e]]   (64 threads per edge)
__global__ void scatter_kernel(const int* __restrict__ seg, const int* __restrict__ gat,
                               const float* __restrict__ fac, const float* __restrict__ tab,
                               float* __restrict__ acc, int nEdges) {
  int t = blockIdx.x * blockDim.x + threadIdx.x;
  int e = t >> 6, d = t & 63;
  if (e < nEdges) {
    int sdst = seg[e], gsrc = gat[e];
    float v = fac[sdst] * tab[(size_t)gsrc * DD + d];
    atomicAdd(&acc[(size_t)sdst * DD + d], v);
  }
}

__global__ void reg_user_kernel(const float* __restrict__ pq_user, const float* __restrict__ yw_user,
                                const float* __restrict__ b_user,
                                const float* __restrict__ Iu, const float* __restrict__ Tu,
                                const float* __restrict__ Tvp,
                                float* __restrict__ acc, int nU) {
  __shared__ float sdata[8];
  int wid = (int)((blockIdx.x * blockDim.x + threadIdx.x) >> 5);
  int lane = threadIdx.x & 31;
  float contrib = 0.0f;
  if (wid < nU) {
    const v2f* pq = (const v2f*)(pq_user + (size_t)wid * DD);
    const v2f* yw = (const v2f*)(yw_user + (size_t)wid * DD);
    v2f p = pq[lane], y = yw[lane];
    float sp = p.x * p.x + p.y * p.y;
    float sy = y.x * y.x + y.y * y.y;
#pragma unroll
    for (int m = 16; m >= 1; m >>= 1) { sp += __shfl_xor(sp, m, 32); sy += __shfl_xor(sy, m, 32); }
    if (lane == 0) {
      float iu = Iu[wid], tu = Tu[wid], tv = Tvp[wid], b = b_user[wid];
      contrib = (LAMDA * iu * b * b + (LAMDA * iu + LAMDA_T * tu) * sp + LAMDA_T * tv * sy)
                / (float)nU;
    }
  }
  float bsum = block_reduce_add(contrib, sdata);
  if (threadIdx.x == 0) atomicAdd(acc, bsum);
}

__global__ void reg_item_kernel(const float* __restrict__ pq_item, const float* __restrict__ yw_item,
                                const float* __restrict__ b_item, const float* __restrict__ Uj,
                                float* __restrict__ acc, int nI) {
  __shared__ float sdata[8];
  int wid = (int)((blockIdx.x * blockDim.x + threadIdx.x) >> 5);
  int lane = threadIdx.x & 31;
  float contrib = 0.0f;
  if (wid < nI) {
    const v2f* pq = (const v2f*)(pq_item + (size_t)wid * DD);
    const v2f* yw = (const v2f*)(yw_item + (size_t)wid * DD);
    v2f p = pq[lane], y = yw[lane];
    float sp = p.x * p.x + p.y * p.y;
    float sy = y.x * y.x + y.y * y.y;
#pragma unroll
    for (int m = 16; m >= 1; m >>= 1) { sp += __shfl_xor(sp, m, 32); sy += __shfl_xor(sy, m, 32); }
    if (lane == 0) {
      float uj = Uj[wid], b = b_item[wid];
      contrib = LAMDA * uj * (b * b + sp + sy) / (float)nI;
    }
  }
  float bsum = block_reduce_add(contrib, sdata);
  if (threadIdx.x == 0) atomicAdd(acc, bsum);
}

__global__ void score_kernel(const float* __restrict__ h_user, const float* __restrict__ h_item,
                             const float* __restrict__ b_user, const float* __restrict__ b_item,
                             const float* __restrict__ gbias,
                             const int* __restrict__ pos_src, const int* __restrict__ pos_dst,
                             const int* __restrict__ neg_src, const int* __restrict__ neg_dst,
                             float* __restrict__ out, int nPred) {
  int wid = (int)((blockIdx.x * blockDim.x + threadIdx.x) >> 5);
  int tilesPer = (nPred + 15) >> 4;
  int which = wid / tilesPer;          // 0 = pos, 1 = neg (wave-uniform)
  if (which >= 2) return;
  int tile = wid - which * tilesPer;
  const int* s = which ? neg_src : pos_src;
  const int* d = which ? neg_dst : pos_dst;
  Dot16 r = tile_dot16(h_user, h_item, s, d, tile * 16, nPred);
  int e = tile * 16 + r.m;
  if (r.active && e < nPred) {
    out[(size_t)which * nPred + e] = r.val + b_user[r.ra] + b_item[r.rb] + gbias[0];
  }
}

__global__ void link_kernel(const float* __restrict__ yw_user, const float* __restrict__ pq_user,
                            const int* __restrict__ ts, const int* __restrict__ td,
                            float* __restrict__ acc, int nT, float scale) {
  __shared__ float sdata[8];
  int wid = (int)((blockIdx.x * blockDim.x + threadIdx.x) >> 5);
  int tiles = (nT + 15) >> 4;
  float contrib = 0.0f;
  if (wid < tiles) {                    // wave-uniform; EXEC full inside
    Dot16 r = tile_dot16(yw_user, pq_user, ts, td, wid * 16, nT);
    int e = wid * 16 + r.m;
    if (r.active && e < nT) {
      float t = r.val - 1.0f;
      contrib = t * t;
    }
  }
  float bsum = block_reduce_add(contrib, sdata);
  if (threadIdx.x == 0) atomicAdd(acc, bsum * scale);
}

__global__ void finalize_kernel(const float* __restrict__ scal, float* __restrict__ out, int off) {
  if (threadIdx.x == 0 && blockIdx.x == 0) {
    out[off]     = scal[0];
    out[off + 1] = scal[1];
  }
}

// ------------------------------------------------------------------- launch

extern "C" void kernel_launch(void* const* d_in, const int* in_sizes, int n_in,
                              void* d_out, int out_size, void* d_ws, size_t ws_size,
                              hipStream_t stream) {
  const float* pq_user = (const float*)d_in[0];
  const float* pq_item = (const float*)d_in[1];
  const float* yw_user = (const float*)d_in[2];
  const float* yw_item = (const float*)d_in[3];
  const float* b_user  = (const float*)d_in[4];
  const float* b_item  = (const float*)d_in[5];
  const float* gbias   = (const float*)d_in[6];
  const int* rate_src  = (const int*)d_in[7];
  const int* rate_dst  = (const int*)d_in[8];
  const int* trust_src = (const int*)d_in[9];
  const int* trust_dst = (const int*)d_in[10];
  const int* pos_src   = (const int*)d_in[11];
  const int* pos_dst   = (const int*)d_in[12];
  const int* neg_src   = (const int*)d_in[13];
  const int* neg_dst   = (const int*)d_in[14];

  const int U_      = in_sizes[4];
  const int I_      = in_sizes[5];
  const int E_RATE  = in_sizes[7];
  const int E_TRUST = in_sizes[9];
  const int E_PRED  = in_sizes[11];

  float* ws = (float*)d_ws;
  float* Iu     = ws;                                  // [U]  deg -> factor in place
  float* Tu     = ws + (size_t)U_;                     // [U]
  float* Tvp    = ws + (size_t)2 * U_;                 // [U]
  float* Uj     = ws + (size_t)3 * U_;                 // [I]
  float* scal   = ws + (size_t)3 * U_ + I_;            // [16] (pads h_user to 16B)
  float* h_user = ws + (size_t)3 * U_ + I_ + 16;       // [U*64]
  float* out    = (float*)d_out;

  const int B = 256;

  // 1) zero degrees + scalar accumulators
  hipMemsetAsync(ws, 0, ((size_t)3 * U_ + I_ + 16) * sizeof(float), stream);

  // 2) degree atomics
  deg2_kernel<<<(E_RATE + B - 1) / B, B, 0, stream>>>(rate_src, Iu, rate_dst, Uj, E_RATE);
  deg2_kernel<<<(E_TRUST + B - 1) / B, B, 0, stream>>>(trust_dst, Tu, trust_src, Tvp, E_TRUST);

  // 3) factors in place (the four arrays are contiguous)
  {
    int n = 3 * U_ + I_;
    factor_kernel<<<(n + B - 1) / B, B, 0, stream>>>(ws, n);
  }

  // 4) h_user = pq_user
  {
    int n4 = (U_ * DD) / 4;
    copy4_kernel<<<(n4 + B - 1) / B, B, 0, stream>>>((const float4*)pq_user, (float4*)h_user, n4);
  }

  // 5) fused scatters: h_user += I_u[s]*yw_item[rate_dst] ; += T_u[d]*yw_user[trust_src]
  {
    long nt = (long)E_RATE * DD;
    scatter_kernel<<<(int)((nt + B - 1) / B), B, 0, stream>>>(rate_src, rate_dst, Iu, yw_item, h_user, E_RATE);
  }
  {
    long nt = (long)E_TRUST * DD;
    scatter_kernel<<<(int)((nt + B - 1) / B), B, 0, stream>>>(trust_dst, trust_src, Tu, yw_user, h_user, E_TRUST);
  }

  // 6) regularization loss
  {
    long nt = (long)U_ * 32;
    reg_user_kernel<<<(int)((nt + B - 1) / B), B, 0, stream>>>(pq_user, yw_user, b_user, Iu, Tu, Tvp, &scal[0], U_);
  }
  {
    long nt = (long)I_ * 32;
    reg_item_kernel<<<(int)((nt + B - 1) / B), B, 0, stream>>>(pq_item, yw_item, b_item, Uj, &scal[0], I_);
  }

  // 7) pos/neg scores (WMMA batched dots, 16 edges/wave)
  {
    int tilesPer = (E_PRED + 15) >> 4;
    long nt = (long)tilesPer * 2 * 32;
    score_kernel<<<(int)((nt + B - 1) / B), B, 0, stream>>>(h_user, pq_item, b_user, b_item, gbias,
                                                            pos_src, pos_dst, neg_src, neg_dst,
                                                            out, E_PRED);
  }

  // 8) link-prediction loss (WMMA batched dots)
  {
    int tiles = (E_TRUST + 15) >> 4;
    long nt = (long)tiles * 32;
    link_kernel<<<(int)((nt + B - 1) / B), B, 0, stream>>>(yw_user, pq_user, trust_src, trust_dst,
                                                           &scal[1], E_TRUST,
                                                           LAMDA_T / (float)E_TRUST);
  }

  // 9) scalars -> output tail
  finalize_kernel<<<1, 1, 0, stream>>>(scal, out, 2 * E_PRED);
}